// GKANModel_72662256714549
// MI455X (gfx1250) — compile-verified
//
#include <hip/hip_runtime.h>
#include <math.h>

// ---------------------------------------------------------------------------
// Types for CDNA5 WMMA (fp32 16x16x4)
// ---------------------------------------------------------------------------
typedef __attribute__((ext_vector_type(2))) float v2f;
typedef __attribute__((ext_vector_type(8))) float v8f;

static __device__ __forceinline__ void atomic_add_f32(float* p, float v) {
  // Lowers to global_atomic_add_f32 on gfx1250 (relaxed, device scope).
  __hip_atomic_fetch_add(p, v, __ATOMIC_RELAXED, __HIP_MEMORY_SCOPE_AGENT);
}

// ---------------------------------------------------------------------------
// deg[i] = 1 (self loop), then += 1 per incoming edge, then dis = rsqrt(deg)
// ---------------------------------------------------------------------------
__global__ void deg_init_kernel(float* __restrict__ deg, int n) {
  int i = blockIdx.x * blockDim.x + threadIdx.x;
  if (i < n) deg[i] = 1.0f;  // self-loop contribution
}

__global__ void deg_count_kernel(const int* __restrict__ col,
                                 float* __restrict__ deg, int e) {
  int i = blockIdx.x * blockDim.x + threadIdx.x;
  if (i < e) atomic_add_f32(&deg[col[i]], 1.0f);
}

__global__ void dis_kernel(float* __restrict__ deg, int n) {
  int i = blockIdx.x * blockDim.x + threadIdx.x;
  if (i < n) {
    float d = deg[i];
    deg[i] = (d > 0.0f) ? rsqrtf(d) : 0.0f;  // in-place: deg -> deg^-1/2
  }
}

// ---------------------------------------------------------------------------
// WMMA GEMM:  H[nrows x ncols] = X[nrows x 64] * W[64 x ncols]
// One wave computes a 16 x (NT*16) tile using V_WMMA_F32_16X16X4_F32.
// W staged in LDS interleaved as [kpair][paddedCol][2] so the per-lane B
// fragment is one 8-byte LDS load.
// ---------------------------------------------------------------------------
template <int NT>
__global__ __launch_bounds__(256) void gemm_wmma_kernel(
    const float* __restrict__ X, const float* __restrict__ W,
    float* __restrict__ H, int nrows, int ncols) {
  constexpr int WC = NT * 16;      // padded output columns
  __shared__ float s_w[64 * WC];   // [32 kpairs][WC cols][2]

  for (int idx = threadIdx.x; idx < 64 * WC; idx += blockDim.x) {
    int k = idx / WC, c = idx % WC;
    float w = (c < ncols) ? W[k * ncols + c] : 0.0f;
    s_w[((k >> 1) * WC + c) * 2 + (k & 1)] = w;
  }
  __syncthreads();

  const int wave = threadIdx.x >> 5;
  const int lane = threadIdx.x & 31;
  const int tile = blockIdx.x * (blockDim.x >> 5) + wave;
  if (tile * 16 >= nrows) return;
  const int row0 = tile * 16;
  const int m  = lane & 15;   // row (A) / col (B,C)
  const int hi = lane >> 4;   // half-wave select

  const v2f* __restrict__ arow = (const v2f*)(X + (size_t)(row0 + m) * 64);
  const v2f* __restrict__ bw   = (const v2f*)s_w;

  v8f acc[NT] = {};

#pragma unroll
  for (int k = 0; k < 16; ++k) {                 // K advances 4 per step
    v2f a = arow[k * 2 + hi];                    // K = 4k + 2*hi, +1
#pragma unroll
    for (int nt = 0; nt < NT; ++nt) {
      v2f b = bw[(k * 2 + hi) * WC + nt * 16 + m];
      acc[nt] = __builtin_amdgcn_wmma_f32_16x16x4_f32(
          false, a, false, b, (short)0, acc[nt], false, false);
    }
  }

#pragma unroll
  for (int nt = 0; nt < NT; ++nt) {
#pragma unroll
    for (int v = 0; v < 8; ++v) {
      int r = row0 + v + hi * 8;                 // C/D layout: VGPR v, half hi
      int c = nt * 16 + m;
      if (c < ncols) H[(size_t)r * ncols + c] = acc[nt][v];
    }
  }
}

// ---------------------------------------------------------------------------
// out[i,c] = bias[c] + h[i,c] * dis[i]^2     (bias + self-loop term)
// ---------------------------------------------------------------------------
__global__ void init_out_kernel(const float* __restrict__ h,
                                const float* __restrict__ dis,
                                const float* __restrict__ bias,
                                float* __restrict__ out, int n, int C) {
  size_t idx = (size_t)blockIdx.x * blockDim.x + threadIdx.x;
  if (idx >= (size_t)n * C) return;
  int i = (int)(idx / C), c = (int)(idx % C);
  float d = dis[i];
  out[idx] = bias[c] + h[idx] * d * d;
}

// ---------------------------------------------------------------------------
// Wave-per-edge scatter: out[col] += h[row] * dis[row]*dis[col]
// ---------------------------------------------------------------------------
__global__ void scatter_kernel(const float* __restrict__ h,
                               const int* __restrict__ rows,
                               const int* __restrict__ cols,
                               const float* __restrict__ dis,
                               float* __restrict__ out, int ne, int C) {
  int w = (blockIdx.x * blockDim.x + threadIdx.x) >> 5;
  int lane = threadIdx.x & 31;
  if (w >= ne) return;
  int r = rows[w], c = cols[w];
  float nrm = dis[r] * dis[c];
  const float* hr = h + (size_t)r * C;
  float* oc = out + (size_t)c * C;
  for (int ch = lane; ch < C; ch += 32)
    atomic_add_f32(oc + ch, hr[ch] * nrm);
}

__global__ void relu_kernel(float* __restrict__ x, size_t n) {
  size_t i = (size_t)blockIdx.x * blockDim.x + threadIdx.x;
  if (i < n) x[i] = fmaxf(x[i], 0.0f);
}

// ---------------------------------------------------------------------------
// Wave-per-row log_softmax over 40 columns (lane and lane+32).
// ---------------------------------------------------------------------------
__global__ void log_softmax_kernel(const float* __restrict__ in,
                                   float* __restrict__ out, int n, int C) {
  int row = (blockIdx.x * blockDim.x + threadIdx.x) >> 5;
  int lane = threadIdx.x & 31;
  if (row >= n) return;
  const float* xr = in + (size_t)row * C;
  bool has2 = (lane + 32) < C;
  float v0 = (lane < C) ? xr[lane] : -INFINITY;
  float v1 = has2 ? xr[lane + 32] : -INFINITY;
  float mx = fmaxf(v0, v1);
#pragma unroll
  for (int off = 16; off > 0; off >>= 1)
    mx = fmaxf(mx, __shfl_xor(mx, off, 32));
  float s = ((lane < C) ? __expf(v0 - mx) : 0.0f) +
            (has2 ? __expf(v1 - mx) : 0.0f);
#pragma unroll
  for (int off = 16; off > 0; off >>= 1)
    s += __shfl_xor(s, off, 32);
  float lse = mx + __logf(s);
  float* orow = out + (size_t)row * C;
  if (lane < C) orow[lane] = v0 - lse;
  if (has2) orow[lane + 32] = v1 - lse;
}

// ---------------------------------------------------------------------------
// Launch
// ---------------------------------------------------------------------------
static inline int cdiv(long long a, long long b) { return (int)((a + b - 1) / b); }

extern "C" void kernel_launch(void* const* d_in, const int* in_sizes, int n_in,
                              void* d_out, int out_size, void* d_ws, size_t ws_size,
                              hipStream_t stream) {
  const float* x  = (const float*)d_in[0];
  const int*   ei = (const int*)d_in[1];
  const float* W1 = (const float*)d_in[2];
  const float* b1 = (const float*)d_in[3];
  const float* W2 = (const float*)d_in[4];
  const float* b2 = (const float*)d_in[5];
  float* out = (float*)d_out;

  const int IN_C = 64, HID_C = 64, OUT_C = 40;
  const int N = in_sizes[0] / IN_C;      // 100000
  const int E = in_sizes[1] / 2;         // 1250000
  const int* row = ei;                   // sources
  const int* col = ei + E;               // targets (aggregation index)

  // Workspace layout (floats). dis:N | h1:N*64 | o1:N*64 | h2:N*40 | o2:N*40
  float* ws  = (float*)d_ws;
  float* dis = ws;
  size_t off = ((size_t)N + 511) & ~(size_t)511;
  float* h1 = ws + off;            off += (size_t)N * HID_C;
  float* o1 = ws + off;            off += (size_t)N * HID_C;
  float* h2 = ws + off;            off += (size_t)N * OUT_C;
  float* o2 = ws + off;

  const int B = 256;

  // Normalization coefficients
  deg_init_kernel<<<cdiv(N, B), B, 0, stream>>>(dis, N);
  deg_count_kernel<<<cdiv(E, B), B, 0, stream>>>(col, dis, E);
  dis_kernel<<<cdiv(N, B), B, 0, stream>>>(dis, N);

  // Layer 1: h1 = x @ W1 (WMMA), o1 = b1 + self + scatter, ReLU
  int rowTiles = cdiv(N, 16);
  int gemmBlocks = cdiv(rowTiles, B / 32);
  gemm_wmma_kernel<4><<<gemmBlocks, B, 0, stream>>>(x, W1, h1, N, HID_C);
  init_out_kernel<<<cdiv((long long)N * HID_C, B), B, 0, stream>>>(h1, dis, b1, o1, N, HID_C);
  scatter_kernel<<<cdiv((long long)E * 32, B), B, 0, stream>>>(h1, row, col, dis, o1, E, HID_C);
  relu_kernel<<<cdiv((long long)N * HID_C, B), B, 0, stream>>>(o1, (size_t)N * HID_C);

  // Layer 2: h2 = relu(o1) @ W2 (WMMA, 40 cols padded to 48), o2 = b2 + self + scatter
  gemm_wmma_kernel<3><<<gemmBlocks, B, 0, stream>>>(o1, W2, h2, N, OUT_C);
  init_out_kernel<<<cdiv((long long)N * OUT_C, B), B, 0, stream>>>(h2, dis, b2, o2, N, OUT_C);
  scatter_kernel<<<cdiv((long long)E * 32, B), B, 0, stream>>>(h2, row, col, dis, o2, E, OUT_C);

  // log_softmax -> d_out
  log_softmax_kernel<<<cdiv((long long)N * 32, B), B, 0, stream>>>(o2, out, N, OUT_C);
}